// Dilation3Dxy_41729902248351
// MI455X (gfx1250) — compile-verified
//
#include <hip/hip_runtime.h>
#include <stdint.h>

// Dilation3Dxy (max-plus 3x3 "SAME" dilation on flattened [B,H,W,C] volume)
//   x      : [8, 128, 32, 32, 8]  -> flat [8, 128, 1024, 8]
//   kernel : [3, 3, 8, 16]        -> flat [9, 8, 16] = 1152 floats
//   out    : [8, 128, 32, 32, 128] (C*F with C outer)
//
// Memory-bound: 512MB streamed out, 32MB in. Strategy: 1 thread per
// (b,y,w,c), 16 filters in registers, NT b128 stores, weights staged to LDS
// via CDNA5 async global->LDS DMA.

#define B_   8
#define H_   128
#define W_   1024
#define C_   8
#define F_   16
#define TAPS 9
#define KELEMS  (TAPS * C_ * F_)   // 1152 floats = 4608 bytes
#define KCHUNKS (KELEMS / 4)       // 288 16-byte chunks

typedef float v4f __attribute__((ext_vector_type(4)));

static __device__ __forceinline__ v4f vmax4(v4f a, v4f b) {
  v4f r;
  r.x = fmaxf(a.x, b.x);
  r.y = fmaxf(a.y, b.y);
  r.z = fmaxf(a.z, b.z);
  r.w = fmaxf(a.w, b.w);
  return r;
}

__global__ __launch_bounds__(256) void dil3dxy_kernel(
    const float* __restrict__ x, const float* __restrict__ kern,
    float* __restrict__ out) {
  __shared__ float sk[KELEMS];

  const int tid = threadIdx.x;

  // ---- Stage the 4.6KB weight tensor into LDS via async DMA --------------
  // Each enabled lane copies one 16B chunk: 288 chunks total. Low 32 bits of
  // the flat address of a __shared__ object are its LDS byte offset (the
  // AS3 pointer value from the addrspacecast), which is exactly what the
  // async instruction's VDST operand expects.
  {
    const uint32_t lds0 = (uint32_t)(uintptr_t)(&sk[0]);
    for (int j = tid; j < KCHUNKS; j += 256) {
      uint32_t lds_addr = lds0 + (uint32_t)j * 16u;
      uint64_t gaddr    = (uint64_t)(uintptr_t)(kern + j * 4);
      asm volatile("global_load_async_to_lds_b128 %0, %1, off"
                   :: "v"(lds_addr), "v"(gaddr)
                   : "memory");
    }
    // Each wave drains its own ASYNCcnt, then the workgroup barrier makes
    // the staged weights visible to every wave.
    asm volatile("s_wait_asynccnt 0" ::: "memory");
    __syncthreads();
  }

  // ---- Per-thread max-plus over 9 taps, 16 filters in registers ----------
  const int i  = blockIdx.x * 256 + tid;   // == flat x index (b,y,w,c)
  const int c  = i & (C_ - 1);
  const int wi = (i >> 3) & (W_ - 1);
  const int y  = (i >> 13) & (H_ - 1);

  const float NEG = -__builtin_inff();
  v4f acc0 = {NEG, NEG, NEG, NEG};
  v4f acc1 = acc0, acc2 = acc0, acc3 = acc0;

#pragma unroll
  for (int t = 0; t < TAPS; ++t) {
    const int dy = t / 3 - 1;
    const int dx = t % 3 - 1;
    // TF 'SAME' pad with -inf on the flattened (H, W=1024) plane.
    const bool valid =
        ((unsigned)(y + dy) < (unsigned)H_) && ((unsigned)(wi + dx) < (unsigned)W_);
    float xv = NEG;
    if (valid) xv = x[i + dy * (W_ * C_) + dx * C_];  // coalesced b32

    // Weight row for (tap t, channel c): 16 floats, 64B-aligned in LDS.
    const v4f* wp = (const v4f*)&sk[t * (C_ * F_) + c * F_];
    v4f xb = {xv, xv, xv, xv};
    acc0 = vmax4(acc0, xb + wp[0]);
    acc1 = vmax4(acc1, xb + wp[1]);
    acc2 = vmax4(acc2, xb + wp[2]);
    acc3 = vmax4(acc3, xb + wp[3]);
  }

  // 64 contiguous bytes per thread; streaming store (don't pollute L2).
  v4f* op = (v4f*)(out + (size_t)i * F_);
  __builtin_nontemporal_store(acc0, op + 0);
  __builtin_nontemporal_store(acc1, op + 1);
  __builtin_nontemporal_store(acc2, op + 2);
  __builtin_nontemporal_store(acc3, op + 3);
}

extern "C" void kernel_launch(void* const* d_in, const int* in_sizes, int n_in,
                              void* d_out, int out_size, void* d_ws, size_t ws_size,
                              hipStream_t stream) {
  const float* x    = (const float*)d_in[0];   // 8*128*32*32*8 f32
  const float* kern = (const float*)d_in[1];   // 3*3*8*16 f32
  float* out        = (float*)d_out;           // 8*128*32*32*128 f32

  const int total  = B_ * H_ * W_ * C_;        // 8,388,608 threads
  const int blocks = total / 256;              // 32,768 blocks
  dil3dxy_kernel<<<dim3(blocks), dim3(256), 0, stream>>>(x, kern, out);
}